// LINKXS_45664092291088
// MI455X (gfx1250) — compile-verified
//
#include <hip/hip_runtime.h>
#include <hip/hip_bf16.h>

typedef __attribute__((ext_vector_type(2))) float v2f;
typedef __attribute__((ext_vector_type(8))) float v8f;

#if defined(__has_builtin)
#if __has_builtin(__builtin_amdgcn_global_load_async_to_lds_b32)
#define HAVE_ASYNC_LDS 1
#endif
#endif
#ifndef HAVE_ASYNC_LDS
#define HAVE_ASYNC_LDS 0
#endif

__device__ __forceinline__ void wait_asynccnt0() {
#if defined(__has_builtin)
#if __has_builtin(__builtin_amdgcn_s_wait_asynccnt)
    __builtin_amdgcn_s_wait_asynccnt(0);
    return;
#else
    asm volatile("s_wait_asynccnt 0x0" ::: "memory");
    return;
#endif
#else
    asm volatile("s_wait_asynccnt 0x0" ::: "memory");
#endif
}

// ---------------------------------------------------------------------------
// Workspace layout (floats):
//   agg   : B*HID   (reused as H after GEMM2 consumes it)
//   HX    : B*HID
//   HA    : B*HID
//   deg   : B
//   in_batch  : Nn ints
//   batch_map : Nn ints
// ---------------------------------------------------------------------------

__global__ void k_init(float* __restrict__ agg, float* __restrict__ deg,
                       int* __restrict__ in_batch, int aggN, int Bsz, int Nn) {
    int i = blockIdx.x * blockDim.x + threadIdx.x;
    if (i < aggN) agg[i] = 0.0f;
    if (i < Bsz)  deg[i] = 0.0f;
    if (i < Nn)   in_batch[i] = 0;
}

__global__ void k_build(const int* __restrict__ batch_nodes,
                        int* __restrict__ in_batch, int* __restrict__ batch_map,
                        int Bsz) {
    int i = blockIdx.x * blockDim.x + threadIdx.x;
    if (i >= Bsz) return;
    int n = batch_nodes[i];
    in_batch[n]  = 1;
    batch_map[n] = i;
}

// One wave per edge: wave gathers W_adj[row] (512B coalesced as float4/lane)
// and atomically accumulates into agg[batch(col)].  agg (51MB) is L2-resident
// on MI455X (192MB L2) so the f32 atomics resolve in L2.
__global__ __launch_bounds__(256) void k_edges(
    const float* __restrict__ W_adj, const int* __restrict__ row,
    const int* __restrict__ col, const int* __restrict__ in_batch,
    const int* __restrict__ batch_map, float* __restrict__ agg,
    float* __restrict__ deg, int E, int HIDc) {
    long long tid = (long long)blockIdx.x * blockDim.x + threadIdx.x;
    int e    = (int)(tid >> 5);
    int lane = (int)(tid & 31);
    if (e >= E) return;
    int c = col[e];
    if (!in_batch[c]) return;
    int b = batch_map[c];
    int r = row[e];
    const float* src = W_adj + (size_t)r * HIDc;
    float*       dst = agg   + (size_t)b * HIDc;
    for (int h = lane * 4; h < HIDc; h += 128) {
        float4 v = *reinterpret_cast<const float4*>(src + h);
        atomicAdd(dst + h + 0, v.x);
        atomicAdd(dst + h + 1, v.y);
        atomicAdd(dst + h + 2, v.z);
        atomicAdd(dst + h + 3, v.w);
    }
    if (lane == 0) atomicAdd(deg + b, 1.0f);
}

// ---------------------------------------------------------------------------
// Unified fp32 WMMA GEMM, double-buffered async LDS staging of B:
//   C[m,n] = act( sum_k scale(m)*Acat[row(m),k]*B[k,n] + bias[n]
//                 (+ add1[m,n] + add2[m,n]) )
// Acat rows: k < K1 -> A[row][k] (stride K1), k >= K1 -> A2[row][k-K1].
// B slab is stored pair-packed: sB[k/2][n] = {B[k][n], B[k+1][n]}, so the
// WMMA B-operand is ONE ds_load_b64 into an even VGPR pair (no re-pairing
// moves).  Row stride chosen so 2*STR2 % 64 == 32: lanes 0-15 / 16-31 hit
// disjoint LDS bank halves.  Slab s+1 is fetched via async global->LDS while
// slab s feeds V_WMMA_F32_16X16X4_F32; s_wait_asynccnt + barrier per slab.
// Requires K % KB == 0 and (A2 ? K1 % KB == 0 : true).
// ---------------------------------------------------------------------------
template <int TN, int KB>
__global__ __launch_bounds__(128) void k_gemm_t(
    const float* __restrict__ A, const float* __restrict__ A2, int K1,
    const int* __restrict__ gather, const float* __restrict__ deg,
    const float* __restrict__ Bm, const float* __restrict__ bias,
    const float* __restrict__ add1, const float* __restrict__ add2,
    float* __restrict__ C, int M, int N, int K, int relu) {
    constexpr int TNC  = TN * 16;               // columns handled by this block
    constexpr int REM  = TNC & 31;
    constexpr int PAD  = (16 - REM) & 31;
    constexpr int STR2 = TNC + PAD;             // float2 row stride: 2*STR2 % 64 == 32
    __shared__ float2 sB[2][(KB / 2) * STR2];

    int wave = threadIdx.x >> 5;
    int lane = threadIdx.x & 31;
    int half = lane >> 4;
    int lm   = lane & 15;
    int tilesM = (M + 15) >> 4;
    int tm = blockIdx.x * 4 + wave;
    bool active = (tm < tilesM);
    if (!active) tm = tilesM - 1;               // clamp: keep all waves for barriers
    int nBase = blockIdx.y * TNC;               // N-group (grid.y covers tilesN>TN)

    int mA = tm * 16 + lm; if (mA >= M) mA = M - 1;     // clamped load row
    int rowA = gather ? gather[mA] : mA;
    float scale = 1.0f;
    if (deg) { float d = fmaxf(deg[mA], 1.0f); scale = 1.0f + 1.0f / d; }
    int ldA  = A2 ? K1 : K;
    const float* Arow  = A + (size_t)rowA * ldA;
    const float* Arow2 = A2 ? (A2 + (size_t)rowA * (size_t)(K - K1)) : Arow;

    // stage slab s (KB rows of B) pair-packed into buffer bufIdx
    auto stage = [&](int s, int bufIdx) {
        int kglob = s * KB;
        float* dstf = reinterpret_cast<float*>(&sB[bufIdx][0]);
        for (int kk = wave; kk < KB; kk += 4) {           // wave-strided rows
            const float* srcRow = Bm + (size_t)(kglob + kk) * N;
            int base = (kk >> 1) * (STR2 * 2) + (kk & 1); // pair row, lo/hi half
            for (int cc = lane; cc < TNC; cc += 32) {     // lane-strided cols (coalesced)
                int ng = nBase + cc; if (ng >= N) ng = N - 1;
#if HAVE_ASYNC_LDS
                __builtin_amdgcn_global_load_async_to_lds_b32(
                    (__attribute__((address_space(1))) int*)(srcRow + ng),
                    (__attribute__((address_space(3))) int*)(dstf + base + cc * 2),
                    0, 0);
#else
                dstf[base + cc * 2] = srcRow[ng];
#endif
            }
        }
    };

    v8f zero = {};
    v8f c[TN];
    #pragma unroll
    for (int t = 0; t < TN; ++t) c[t] = zero;

    const int S = K / KB;
    stage(0, 0);
#if HAVE_ASYNC_LDS
    wait_asynccnt0();
#endif
    __syncthreads();

    for (int s = 0; s < S; ++s) {
        if (s + 1 < S) stage(s + 1, (s + 1) & 1);        // overlap: fill other buffer
        int kglob = s * KB;
        const float* Asrc = (kglob < K1) ? (Arow + kglob) : (Arow2 + (kglob - K1));
        __builtin_prefetch((const void*)(Asrc + KB), 0, 3);   // next A slab, WGP scope
        const float2* sbuf = &sB[s & 1][0];
        for (int k0 = 0; k0 < KB; k0 += 4) {
            int kl = k0 + 2 * half;                       // kl, kl+1 for vgpr 0,1
            float2 a2 = *reinterpret_cast<const float2*>(Asrc + kl);
            v2f a; a[0] = a2.x * scale; a[1] = a2.y * scale;
            const float2* sp = sbuf + ((k0 >> 1) + half) * STR2;
            #pragma unroll
            for (int t = 0; t < TN; ++t) {
                float2 bb = sp[t * 16 + lm];              // one ds_load_b64
                v2f b; b[0] = bb.x; b[1] = bb.y;
                c[t] = __builtin_amdgcn_wmma_f32_16x16x4_f32(false, a, false, b,
                                                             (short)0, c[t], false, false);
            }
        }
#if HAVE_ASYNC_LDS
        wait_asynccnt0();                                 // slab s+1 landed
#endif
        __syncthreads();                                  // all waves done with buf s
    }

    #pragma unroll
    for (int t = 0; t < TN; ++t) {
        int n = nBase + t * 16 + lm;
        if (n < N) {
            float bn = bias[n];
            #pragma unroll
            for (int r = 0; r < 8; ++r) {
                int m = tm * 16 + r + 8 * half;
                if (active && m < M) {
                    size_t idx = (size_t)m * N + n;
                    float v = c[t][r] + bn;
                    if (add1) v += add1[idx];
                    if (add2) v += add2[idx];
                    if (relu) v = fmaxf(v, 0.0f);
                    C[idx] = v;
                }
            }
        }
    }
}

extern "C" void kernel_launch(void* const* d_in, const int* in_sizes, int n_in,
                              void* d_out, int out_size, void* d_ws, size_t ws_size,
                              hipStream_t stream) {
    const float* X           = (const float*)d_in[0];
    const int*   edge_index  = (const int*)d_in[1];
    const int*   batch_nodes = (const int*)d_in[2];
    const float* W_adj       = (const float*)d_in[3];
    const float* W1          = (const float*)d_in[4];
    const float* b1          = (const float*)d_in[5];
    const float* W2          = (const float*)d_in[6];
    const float* b2          = (const float*)d_in[7];
    const float* Ww          = (const float*)d_in[8];
    const float* bw          = (const float*)d_in[9];
    const float* Wo          = (const float*)d_in[10];
    const float* bo          = (const float*)d_in[11];
    float*       out         = (float*)d_out;

    const int Bsz  = in_sizes[2];               // batch size (== N here)
    const int HID  = in_sizes[5];               // b1 length
    const int FEAT = in_sizes[4] / HID;         // W1 is FEAT x HID
    const int NCLS = in_sizes[11];              // bo length
    const int Nn   = in_sizes[3] / HID;         // W_adj is Nn x HID
    const int E    = in_sizes[1] / 2;           // edge_index is 2 x E

    const int* erow = edge_index;
    const int* ecol = edge_index + E;

    float* ws  = (float*)d_ws;
    float* agg = ws;                                  // Bsz*HID (reused as H)
    float* HX  = agg + (size_t)Bsz * HID;
    float* HA  = HX  + (size_t)Bsz * HID;
    float* deg = HA  + (size_t)Bsz * HID;             // Bsz
    int*   in_batch  = (int*)(deg + Bsz);             // Nn
    int*   batch_map = in_batch + Nn;                 // Nn

    const int aggN    = Bsz * HID;
    const int blocksM = (Bsz + 63) / 64;              // 4 waves (=4 M-tiles) / block

    // 1) zero agg/deg/in_batch (ws is poisoned before timing)
    {
        int threads = 256, blocks = (aggN + threads - 1) / threads;
        k_init<<<blocks, threads, 0, stream>>>(agg, deg, in_batch, aggN, Bsz, Nn);
    }
    // 2) batch membership / index maps
    {
        int threads = 256, blocks = (Bsz + threads - 1) / threads;
        k_build<<<blocks, threads, 0, stream>>>(batch_nodes, in_batch, batch_map, Bsz);
    }
    // 3) edge scatter-add (dominant memory phase)
    {
        long long total = (long long)E * 32;
        int threads = 256;
        int blocks = (int)((total + threads - 1) / threads);
        k_edges<<<blocks, threads, 0, stream>>>(W_adj, erow, ecol, in_batch,
                                                batch_map, agg, deg, E, HID);
    }
    // 4) HX = relu(X[batch] @ W1 + b1)          (N=128 -> TN=8, grid.y=1)
    {
        int tilesN = (HID + 15) / 16;
        dim3 grid(blocksM, (tilesN + 7) / 8);
        k_gemm_t<8, 32><<<grid, 128, 0, stream>>>(
            X, nullptr, FEAT, batch_nodes, nullptr, W1, b1,
            nullptr, nullptr, HX, Bsz, HID, FEAT, 1);
    }
    // 5) HA = relu( (agg*(1+1/max(deg,1))) @ W2 + b2 ) — degree scale fused in A-load
    {
        int tilesN = (HID + 15) / 16;
        dim3 grid(blocksM, (tilesN + 7) / 8);
        k_gemm_t<8, 32><<<grid, 128, 0, stream>>>(
            agg, nullptr, HID, nullptr, deg, W2, b2,
            nullptr, nullptr, HA, Bsz, HID, HID, 1);
    }
    // 6) H = concat(HX,HA) @ Ww + bw + HX + HA   (H overwrites agg, now dead)
    {
        int tilesN = (HID + 15) / 16;
        dim3 grid(blocksM, (tilesN + 7) / 8);
        k_gemm_t<8, 32><<<grid, 128, 0, stream>>>(
            HX, HA, HID, nullptr, nullptr, Ww, bw,
            HX, HA, agg, Bsz, HID, 2 * HID, 0);
    }
    // 7) out = H @ Wo + bo   (N=40 -> 3 N-tiles, TN=3, partial tile guarded)
    {
        int tilesN = (NCLS + 15) / 16;
        dim3 grid(blocksM, (tilesN + 2) / 3);
        k_gemm_t<3, 32><<<grid, 128, 0, stream>>>(
            agg, nullptr, HID, nullptr, nullptr, Wo, bo,
            nullptr, nullptr, out, Bsz, NCLS, HID, 0);
    }
}